// MemoryProjection_52956946760024
// MI455X (gfx1250) — compile-verified
//
#include <hip/hip_runtime.h>

typedef __attribute__((ext_vector_type(2))) float v2f;
typedef __attribute__((ext_vector_type(8))) float v8f;

#define L_STEPS 2048
#define NSEQ    512      // BATCH * SIZE
#define N_ORD   64
#define T_CHUNK 64       // timesteps per chunk
#define NCHUNK  (L_STEPS / T_CHUNK)   // 32
#define NGRP    (NSEQ / 16)           // 32 sequence groups (16 seqs per wave)
#define NPOW    6        // A^(2^6) = A^64

// ---------------------------------------------------------------------------
// Shared wave-level machinery (wave32; WMMA f32 16x16x4).
//
// State X is 64(n) x 16(seq) per wave.  A-operand fragments: 16x4 f32 tiles,
// lane<16 holds A[M=lane][k..k+1], lane>=16 holds A[M=lane-16][k+2..k+3].
// B-operand fragments: 4x16 tiles, v0 = X[kb+0|kb+2][col], v1 = X[kb+1|kb+3].
// C/D: VGPR r, lane<16 -> row r, lane>=16 -> row r+8; col = lane&15.
// ---------------------------------------------------------------------------

__device__ __forceinline__ void load_afrag(const float* __restrict__ M,
                                           int lrow, int half,
                                           v2f af[4][4][4]) {
#pragma unroll
    for (int nt = 0; nt < 4; ++nt)
#pragma unroll
        for (int m = 0; m < 4; ++m)
#pragma unroll
            for (int kc = 0; kc < 4; ++kc)
                af[nt][m][kc] = *(const v2f*)(M + (nt * 16 + lrow) * N_ORD +
                                              m * 16 + kc * 4 + half * 2);
}

// acc(64x16) += M(64x64) @ X(64x16): 64 WMMAs, 4 independent chains (nt).
__device__ __forceinline__ void mm_step(const v2f af[4][4][4],
                                        const v2f xb[4][4], v8f acc[4]) {
#pragma unroll
    for (int m = 0; m < 4; ++m)
#pragma unroll
        for (int kc = 0; kc < 4; ++kc) {
            const v2f b = xb[m][kc];
#pragma unroll
            for (int nt = 0; nt < 4; ++nt)
                acc[nt] = __builtin_amdgcn_wmma_f32_16x16x4_f32(
                    false, af[nt][m][kc], false, b, (short)0, acc[nt],
                    false, false);
        }
}

// C/D-layout tile -> 4 B-operand fragments (half-wave swap + select).
__device__ __forceinline__ void relayout(const v8f d, bool lo, v2f xbm[4]) {
    float s[8];
#pragma unroll
    for (int r = 0; r < 8; ++r)
        s[r] = __shfl_xor(d[r], 16, 32);
    xbm[0].x = lo ? d[0] : s[2];  xbm[0].y = lo ? d[1] : s[3];  // rows 0..3
    xbm[1].x = lo ? d[4] : s[6];  xbm[1].y = lo ? d[5] : s[7];  // rows 4..7
    xbm[2].x = lo ? s[0] : d[2];  xbm[2].y = lo ? s[1] : d[3];  // rows 8..11
    xbm[3].x = lo ? s[4] : d[6];  xbm[3].y = lo ? s[5] : d[7];  // rows 12..15
}

// ---------------------------------------------------------------------------
// P0: one workgroup computes A^64 by 6 WMMA squarings staged through LDS.
// Wave w owns output columns 16w..16w+15 of each product.
// ---------------------------------------------------------------------------
__global__ __launch_bounds__(128, 1)
void p0_apow(const float* __restrict__ A, float* __restrict__ A64) {
    __shared__ float buf[2][N_ORD * N_ORD];   // 2 x 16 KB ping-pong

    const int tid  = threadIdx.x;
    const int lane = tid & 31;
    const int w    = tid >> 5;                // wave id 0..3
    const int half = lane >> 4;
    const int lrow = lane & 15;
    const int col  = w * 16 + lrow;           // this wave's output column

    // load A into buf[0] (coalesced)
#pragma unroll
    for (int i = 0; i < 32; ++i)
        buf[0][tid + i * 128] = A[tid + i * 128];
    __syncthreads();

    int p = 0;
    for (int it = 0; it < NPOW; ++it) {
        // A-operand fragments of M from LDS
        v2f af[4][4][4];
#pragma unroll
        for (int nt = 0; nt < 4; ++nt)
#pragma unroll
            for (int m = 0; m < 4; ++m)
#pragma unroll
                for (int kc = 0; kc < 4; ++kc)
                    af[nt][m][kc] = *(const v2f*)&buf[p][(nt * 16 + lrow) * N_ORD +
                                                        m * 16 + kc * 4 + half * 2];
        // B-operand fragments of M (this wave's 16 columns)
        v2f xb[4][4];
#pragma unroll
        for (int m = 0; m < 4; ++m)
#pragma unroll
            for (int kc = 0; kc < 4; ++kc) {
                const int kb = m * 16 + kc * 4 + half * 2;
                xb[m][kc].x = buf[p][(kb + 0) * N_ORD + col];
                xb[m][kc].y = buf[p][(kb + 1) * N_ORD + col];
            }

        v8f acc[4];
#pragma unroll
        for (int nt = 0; nt < 4; ++nt)
            acc[nt] = (v8f)(0.0f);
        mm_step(af, xb, acc);

        // scatter C/D layout into buf[p^1]
#pragma unroll
        for (int nt = 0; nt < 4; ++nt)
#pragma unroll
            for (int r = 0; r < 8; ++r)
                buf[p ^ 1][(nt * 16 + half * 8 + r) * N_ORD + col] = acc[nt][r];
        __syncthreads();
        p ^= 1;
    }

    // NPOW even -> result sits in buf[0]
#pragma unroll
    for (int i = 0; i < 32; ++i)
        A64[tid + i * 128] = buf[p][tid + i * 128];
}

// ---------------------------------------------------------------------------
// P1: local scans. Wave (g,c) runs T_CHUNK steps from x=0, stores only its
// final state (as [16 seq][64 n]) to sLoc.  1024 independent waves.
// ---------------------------------------------------------------------------
__global__ __launch_bounds__(32, 1)
void p1_local(const float* __restrict__ in, const float* __restrict__ A,
              const float* __restrict__ Bm, float* __restrict__ sLoc) {
    const int lane = threadIdx.x & 31;
    const int half = lane >> 4;
    const int lrow = lane & 15;
    const bool lo  = (half == 0);
    const int g = blockIdx.x, c = blockIdx.y;

    v2f af[4][4][4];
    load_afrag(A, lrow, half, af);
    v8f bmv[4];
#pragma unroll
    for (int nt = 0; nt < 4; ++nt)
        bmv[nt] = *(const v8f*)(Bm + nt * 16 + half * 8);

    v2f xb[4][4];
#pragma unroll
    for (int m = 0; m < 4; ++m)
#pragma unroll
        for (int kc = 0; kc < 4; ++kc)
            xb[m][kc] = (v2f)(0.0f);

    const float* ip = in + (size_t)c * T_CHUNK * NSEQ + g * 16 + lrow;
    float inv_next = ip[0];

    v8f acc[4];
    for (int t = 0; t < T_CHUNK; ++t) {
        const float inv = inv_next;
        const int tn = (t + 1 < T_CHUNK) ? (t + 1) : t;
        inv_next = ip[(size_t)tn * NSEQ];

#pragma unroll
        for (int nt = 0; nt < 4; ++nt)
            acc[nt] = bmv[nt] * inv;
        mm_step(af, xb, acc);
#pragma unroll
        for (int m = 0; m < 4; ++m)
            relayout(acc[m], lo, xb[m]);
    }

    float* sp = sLoc + ((size_t)c * NGRP + g) * (16 * N_ORD) + lrow * N_ORD + half * 8;
#pragma unroll
    for (int nt = 0; nt < 4; ++nt)
        *(v8f*)(sp + nt * 16) = acc[nt];
}

// ---------------------------------------------------------------------------
// P2: chunk-state scan. Wave g: s(c) = local(c) + A64 @ s(c-1); records each
// chunk's true start state s(c-1) before computing s(c).  Depth NCHUNK=32.
// ---------------------------------------------------------------------------
__global__ __launch_bounds__(32, 1)
void p2_combine(const float* __restrict__ A64, const float* __restrict__ sLoc,
                float* __restrict__ sStart) {
    const int lane = threadIdx.x & 31;
    const int half = lane >> 4;
    const int lrow = lane & 15;
    const bool lo  = (half == 0);
    const int g = blockIdx.x;

    v2f af[4][4][4];
    load_afrag(A64, lrow, half, af);

    v2f xb[4][4];
#pragma unroll
    for (int m = 0; m < 4; ++m)
#pragma unroll
        for (int kc = 0; kc < 4; ++kc)
            xb[m][kc] = (v2f)(0.0f);

    v8f prev[4];                              // s(c-1) in C/D layout
#pragma unroll
    for (int nt = 0; nt < 4; ++nt)
        prev[nt] = (v8f)(0.0f);

    for (int c = 0; c < NCHUNK; ++c) {
        const size_t off = ((size_t)c * NGRP + g) * (16 * N_ORD) +
                           lrow * N_ORD + half * 8;
        float* stp       = sStart + off;
        const float* slp = sLoc + off;

        v8f acc[4];
#pragma unroll
        for (int nt = 0; nt < 4; ++nt) {
            *(v8f*)(stp + nt * 16) = prev[nt];          // start state of chunk c
            acc[nt] = *(const v8f*)(slp + nt * 16);     // local(c)
        }
        mm_step(af, xb, acc);                           // += A64 @ s(c-1)
#pragma unroll
        for (int nt = 0; nt < 4; ++nt)
            prev[nt] = acc[nt];
#pragma unroll
        for (int m = 0; m < 4; ++m)
            relayout(acc[m], lo, xb[m]);
    }
}

// ---------------------------------------------------------------------------
// P3: final scans. Wave (g,c) reloads its true start state, reruns T_CHUNK
// steps and writes every output x_t.  1024 independent waves.
// ---------------------------------------------------------------------------
__global__ __launch_bounds__(32, 1)
void p3_emit(const float* __restrict__ in, const float* __restrict__ A,
             const float* __restrict__ Bm, const float* __restrict__ sStart,
             float* __restrict__ out) {
    const int lane = threadIdx.x & 31;
    const int half = lane >> 4;
    const int lrow = lane & 15;
    const bool lo  = (half == 0);
    const int g = blockIdx.x, c = blockIdx.y;

    v2f af[4][4][4];
    load_afrag(A, lrow, half, af);
    v8f bmv[4];
#pragma unroll
    for (int nt = 0; nt < 4; ++nt)
        bmv[nt] = *(const v8f*)(Bm + nt * 16 + half * 8);

    // init state fragments from sStart ([16 seq][64 n] -> contiguous v2f pairs)
    const float* sp0 = sStart + ((size_t)c * NGRP + g) * (16 * N_ORD) + lrow * N_ORD;
    v2f xb[4][4];
#pragma unroll
    for (int m = 0; m < 4; ++m)
#pragma unroll
        for (int kc = 0; kc < 4; ++kc)
            xb[m][kc] = *(const v2f*)(sp0 + m * 16 + kc * 4 + half * 2);

    const float* ip = in + (size_t)c * T_CHUNK * NSEQ + g * 16 + lrow;
    float* op = out + (size_t)c * T_CHUNK * NSEQ * N_ORD +
                (size_t)(g * 16 + lrow) * N_ORD + half * 8;
    float inv_next = ip[0];

    for (int t = 0; t < T_CHUNK; ++t) {
        const float inv = inv_next;
        const int tn = (t + 1 < T_CHUNK) ? (t + 1) : t;
        inv_next = ip[(size_t)tn * NSEQ];

        v8f acc[4];
#pragma unroll
        for (int nt = 0; nt < 4; ++nt)
            acc[nt] = bmv[nt] * inv;
        mm_step(af, xb, acc);

        float* o = op + (size_t)t * (NSEQ * N_ORD);
#pragma unroll
        for (int nt = 0; nt < 4; ++nt)
            *(v8f*)(o + nt * 16) = acc[nt];

#pragma unroll
        for (int m = 0; m < 4; ++m)
            relayout(acc[m], lo, xb[m]);
    }
}

// ---------------------------------------------------------------------------
// Fallback: proven monolithic scan (round-1 kernel), used if d_ws is small.
// ---------------------------------------------------------------------------
__global__ __launch_bounds__(32, 1)
void hippo_scan_mono(const float* __restrict__ in, const float* __restrict__ A,
                     const float* __restrict__ Bm, float* __restrict__ out) {
    const int lane = threadIdx.x & 31;
    const int half = lane >> 4;
    const int lrow = lane & 15;
    const bool lo  = (half == 0);
    const int seq  = blockIdx.x * 16 + lrow;

    v2f af[4][4][4];
    load_afrag(A, lrow, half, af);
    v8f bmv[4];
#pragma unroll
    for (int nt = 0; nt < 4; ++nt)
        bmv[nt] = *(const v8f*)(Bm + nt * 16 + half * 8);
    v2f xb[4][4];
#pragma unroll
    for (int m = 0; m < 4; ++m)
#pragma unroll
        for (int kc = 0; kc < 4; ++kc)
            xb[m][kc] = (v2f)(0.0f);

    const float* ip = in + seq;
    float* op = out + (size_t)seq * N_ORD + half * 8;
    float inv_next = ip[0];

    for (int t = 0; t < L_STEPS; ++t) {
        const float inv = inv_next;
        const int tn = (t + 1 < L_STEPS) ? (t + 1) : t;
        inv_next = ip[(size_t)tn * NSEQ];

        v8f acc[4];
#pragma unroll
        for (int nt = 0; nt < 4; ++nt)
            acc[nt] = bmv[nt] * inv;
        mm_step(af, xb, acc);

        float* o = op + (size_t)t * (NSEQ * N_ORD);
#pragma unroll
        for (int nt = 0; nt < 4; ++nt)
            *(v8f*)(o + nt * 16) = acc[nt];
#pragma unroll
        for (int m = 0; m < 4; ++m)
            relayout(acc[m], lo, xb[m]);
    }
}

extern "C" void kernel_launch(void* const* d_in, const int* in_sizes, int n_in,
                              void* d_out, int out_size, void* d_ws, size_t ws_size,
                              hipStream_t stream) {
    const float* in = (const float*)d_in[0];   // (2048, 8, 64) f32
    const float* A  = (const float*)d_in[1];   // (64, 64) f32
    const float* Bm = (const float*)d_in[2];   // (64, 1) f32
    float* out = (float*)d_out;                // (2048, 8, 64, 64) f32

    const size_t stateFloats = (size_t)NCHUNK * NGRP * 16 * N_ORD;  // 1M floats
    const size_t needBytes = (4096 + 2 * stateFloats) * sizeof(float); // ~8.4 MB

    if (ws_size >= needBytes) {
        float* A64    = (float*)d_ws;
        float* sLoc   = A64 + 4096;
        float* sStart = sLoc + stateFloats;

        p0_apow   <<<dim3(1),            dim3(128), 0, stream>>>(A, A64);
        p1_local  <<<dim3(NGRP, NCHUNK), dim3(32),  0, stream>>>(in, A, Bm, sLoc);
        p2_combine<<<dim3(NGRP),         dim3(32),  0, stream>>>(A64, sLoc, sStart);
        p3_emit   <<<dim3(NGRP, NCHUNK), dim3(32),  0, stream>>>(in, A, Bm, sStart, out);
    } else {
        hippo_scan_mono<<<dim3(NSEQ / 16), dim3(32), 0, stream>>>(in, A, Bm, out);
    }
}